// GCNOnly_50130858279695
// MI455X (gfx1250) — compile-verified
//
#include <hip/hip_runtime.h>

#define T_    20
#define B_    4
#define NN_   500
#define BN_   2000
#define DIN_  2
#define DH_   256
#define DOUT_ 128

typedef __attribute__((ext_vector_type(2))) float v2f;
typedef __attribute__((ext_vector_type(8))) float v8f;

// ---------------------------------------------------------------------------
// CDNA5 async global->LDS copies (ASYNCcnt-tracked DMA), per ISA §15.18.3
// ---------------------------------------------------------------------------
__device__ __forceinline__ void async_b128(void* lds, const void* g) {
  unsigned l = (unsigned)(uintptr_t)lds;
  asm volatile("global_load_async_to_lds_b128 %0, %1, off" :: "v"(l), "v"(g) : "memory");
}
__device__ __forceinline__ void async_b32(void* lds, const void* g) {
  unsigned l = (unsigned)(uintptr_t)lds;
  asm volatile("global_load_async_to_lds_b32 %0, %1, off" :: "v"(l), "v"(g) : "memory");
}
#define WAIT_ASYNC(n) asm volatile("s_wait_asynccnt %0" :: "n"(n) : "memory")

// ---------------------------------------------------------------------------
// mask [B,T,N] (int) -> fm [T, B*N] (float)
// ---------------------------------------------------------------------------
__global__ void k_mask(const int* __restrict__ mask, float* __restrict__ fm) {
  int idx = blockIdx.x * blockDim.x + threadIdx.x;
  if (idx >= B_ * T_ * NN_) return;
  int b = idx / (T_ * NN_);
  int rem = idx - b * (T_ * NN_);
  int t = rem / NN_;
  int n = rem - t * NN_;
  fm[t * BN_ + b * NN_ + n] = (float)mask[idx];
}

// ---------------------------------------------------------------------------
// dinv[t,i] = m_i ? rsqrt(1 + sum_j A[t,i,j]*m_j) : 0   (one wave32 per row)
// ---------------------------------------------------------------------------
__global__ void k_dinv(const float* __restrict__ A, const float* __restrict__ fm,
                       float* __restrict__ dinv) {
  int wid = blockIdx.x * (blockDim.x >> 5) + (threadIdx.x >> 5);
  int lane = threadIdx.x & 31;
  if (wid >= T_ * BN_) return;
  int t = wid / BN_;
  int j = wid - t * BN_;
  const float* Arow = A + (size_t)t * BN_ * BN_ + (size_t)j * BN_;
  const float* mt = fm + t * BN_;
  float s = 0.f;
  for (int c = lane; c < BN_; c += 32) s += Arow[c] * mt[c];
#pragma unroll
  for (int off = 16; off > 0; off >>= 1) s += __shfl_xor(s, off, 32);
  if (lane == 0) {
    float mi = fm[wid];
    dinv[wid] = (mi > 0.f) ? rsqrtf(1.f + s) : 0.f;
  }
}

// ---------------------------------------------------------------------------
// hw1[i,h] = (x[i,0]*W1[0,h] + x[i,1]*W1[1,h]) * dinv[i]
// ---------------------------------------------------------------------------
__global__ void k_hw1(const float* __restrict__ x, const float* __restrict__ W1,
                      const float* __restrict__ dinv, float* __restrict__ out) {
  int row = blockIdx.x;
  int h = threadIdx.x;
  float x0 = x[row * 2 + 0], x1 = x[row * 2 + 1];
  out[(size_t)row * DH_ + h] = (x0 * W1[h] + x1 * W1[DH_ + h]) * dinv[row];
}

// ---------------------------------------------------------------------------
// Propagation: Hout = relu( dinv_i * (A[t] @ HW + HW_self) + bias )
// 256 threads (8 waves); tile = 16 rows x 256 cols (each wave: cols n0, n0+128)
// Double-buffered LDS, async global->LDS staging.
// ---------------------------------------------------------------------------
__global__ void __launch_bounds__(256) k_prop(
    const float* __restrict__ A, const float* __restrict__ HW,
    const float* __restrict__ dinv, const float* __restrict__ bias,
    float* __restrict__ Hout) {
  const int t = blockIdx.y;
  const int row0 = blockIdx.x * 16;
  const int tid = threadIdx.x;
  const int wave = tid >> 5, lane = tid & 31;
  const int n0 = wave * 16;
  __shared__ float As[2][16][20];
  __shared__ float Bs[2][16][272];
  const float* Arow = A + (size_t)t * BN_ * BN_ + (size_t)row0 * BN_;
  const float* HWt = HW + (size_t)t * BN_ * DH_;
  const int m = lane & 15;
  const int kb = (lane >> 4) << 1;        // K pair offset: 0 or 2
  // per-thread staging assignments
  const int ar = tid >> 4, ac = tid & 15;           // A tile: 1 b32 / thread
  v8f acc0 = {}, acc1 = {};

  // prologue: issue buffer 0 (k0 = 0)
  async_b32(&As[0][ar][ac], Arow + (size_t)ar * BN_ + ac);
#pragma unroll
  for (int q = 0; q < 4; ++q) {
    int idx4 = tid + q * 256;
    int r = idx4 >> 6, c = (idx4 & 63) << 2;
    async_b128(&Bs[0][r][c], HWt + (size_t)r * DH_ + c);
  }

  for (int k0 = 0; k0 < BN_; k0 += 16) {
    const int cur = (k0 >> 4) & 1;
    if (k0 + 16 < BN_) {
      const int nxt = cur ^ 1;
      async_b32(&As[nxt][ar][ac], Arow + (size_t)ar * BN_ + (k0 + 16) + ac);
#pragma unroll
      for (int q = 0; q < 4; ++q) {
        int idx4 = tid + q * 256;
        int r = idx4 >> 6, c = (idx4 & 63) << 2;
        async_b128(&Bs[nxt][r][c], HWt + (size_t)(k0 + 16 + r) * DH_ + c);
      }
      WAIT_ASYNC(5);     // previous buffer's 5 async ops done (in-order)
    } else {
      WAIT_ASYNC(0);
    }
    __syncthreads();     // all waves' staging for `cur` visible
#pragma unroll
    for (int kk = 0; kk < 16; kk += 4) {
      v2f a, b0, b1;
      a.x = As[cur][m][kk + kb];            a.y = As[cur][m][kk + kb + 1];
      b0.x = Bs[cur][kk + kb][n0 + m];      b0.y = Bs[cur][kk + kb + 1][n0 + m];
      b1.x = Bs[cur][kk + kb][n0 + 128 + m];
      b1.y = Bs[cur][kk + kb + 1][n0 + 128 + m];
      acc0 = __builtin_amdgcn_wmma_f32_16x16x4_f32(false, a, false, b0,
                                                   (short)0, acc0, false, false);
      acc1 = __builtin_amdgcn_wmma_f32_16x16x4_f32(false, a, false, b1,
                                                   (short)0, acc1, false, false);
    }
    __syncthreads();     // safe to overwrite this buffer next iteration
  }

  const int col = n0 + m;
  const int mhi = (lane >> 4) << 3;         // 0 or 8
  const float bv0 = bias[col], bv1 = bias[col + 128];
#pragma unroll
  for (int r = 0; r < 8; ++r) {
    int row = row0 + r + mhi;
    size_t gi = (size_t)t * BN_ + row;
    float di = dinv[gi];
    float sv0 = HWt[(size_t)row * DH_ + col];          // self-loop terms
    float sv1 = HWt[(size_t)row * DH_ + col + 128];
    Hout[gi * DH_ + col]       = fmaxf(di * (acc0[r] + sv0) + bv0, 0.f);
    Hout[gi * DH_ + col + 128] = fmaxf(di * (acc1[r] + sv1) + bv1, 0.f);
  }
}

// ---------------------------------------------------------------------------
// hw2 = (H @ W[256x256]) * dinv ; same tiling/async scheme, K = 256
// ---------------------------------------------------------------------------
__global__ void __launch_bounds__(256) k_mm_scale(
    const float* __restrict__ H, const float* __restrict__ W,
    const float* __restrict__ dinv, float* __restrict__ out) {
  const int row0 = blockIdx.x * 16;
  const int tid = threadIdx.x;
  const int wave = tid >> 5, lane = tid & 31;
  const int n0 = wave * 16;
  __shared__ float As[2][16][20];
  __shared__ float Bs[2][16][272];
  const int m = lane & 15, kb = (lane >> 4) << 1;
  const int ar = tid >> 4, ac = tid & 15;
  v8f acc0 = {}, acc1 = {};

  async_b32(&As[0][ar][ac], H + (size_t)(row0 + ar) * DH_ + ac);
#pragma unroll
  for (int q = 0; q < 4; ++q) {
    int idx4 = tid + q * 256;
    int r = idx4 >> 6, c = (idx4 & 63) << 2;
    async_b128(&Bs[0][r][c], W + (size_t)r * DH_ + c);
  }

  for (int k0 = 0; k0 < DH_; k0 += 16) {
    const int cur = (k0 >> 4) & 1;
    if (k0 + 16 < DH_) {
      const int nxt = cur ^ 1;
      async_b32(&As[nxt][ar][ac], H + (size_t)(row0 + ar) * DH_ + (k0 + 16) + ac);
#pragma unroll
      for (int q = 0; q < 4; ++q) {
        int idx4 = tid + q * 256;
        int r = idx4 >> 6, c = (idx4 & 63) << 2;
        async_b128(&Bs[nxt][r][c], W + (size_t)(k0 + 16 + r) * DH_ + c);
      }
      WAIT_ASYNC(5);
    } else {
      WAIT_ASYNC(0);
    }
    __syncthreads();
#pragma unroll
    for (int kk = 0; kk < 16; kk += 4) {
      v2f a, b0, b1;
      a.x = As[cur][m][kk + kb];            a.y = As[cur][m][kk + kb + 1];
      b0.x = Bs[cur][kk + kb][n0 + m];      b0.y = Bs[cur][kk + kb + 1][n0 + m];
      b1.x = Bs[cur][kk + kb][n0 + 128 + m];
      b1.y = Bs[cur][kk + kb + 1][n0 + 128 + m];
      acc0 = __builtin_amdgcn_wmma_f32_16x16x4_f32(false, a, false, b0,
                                                   (short)0, acc0, false, false);
      acc1 = __builtin_amdgcn_wmma_f32_16x16x4_f32(false, a, false, b1,
                                                   (short)0, acc1, false, false);
    }
    __syncthreads();
  }

  const int col = n0 + m;
  const int mhi = (lane >> 4) << 3;
#pragma unroll
  for (int r = 0; r < 8; ++r) {
    int row = row0 + r + mhi;
    float di = dinv[row];
    out[(size_t)row * DH_ + col]       = di * acc0[r];
    out[(size_t)row * DH_ + col + 128] = di * acc1[r];
  }
}

// ---------------------------------------------------------------------------
// out = (H @ Wf[256x128] + bf) * m, permuted store to [B, N, T, DOUT]
// 256 threads (8 waves), 16x128 tile; synchronous staging (cheap kernel).
// ---------------------------------------------------------------------------
__global__ void __launch_bounds__(256) k_final(
    const float* __restrict__ H, const float* __restrict__ Wf,
    const float* __restrict__ bf, const float* __restrict__ fm,
    float* __restrict__ out) {
  const int row0 = blockIdx.x * 16;
  const int tid = threadIdx.x;
  const int wave = tid >> 5, lane = tid & 31;
  const int n0 = wave * 16;
  __shared__ float As[16][20];
  __shared__ float Bs[16][136];
  const int m = lane & 15, kb = (lane >> 4) << 1;
  v8f acc = {};
  for (int k0 = 0; k0 < DH_; k0 += 16) {
    __syncthreads();
    As[tid >> 4][tid & 15] = H[(size_t)(row0 + (tid >> 4)) * DH_ + k0 + (tid & 15)];
#pragma unroll
    for (int q = 0; q < 2; ++q) {
      int idx4 = tid + q * 256;
      int r = idx4 >> 5, c = (idx4 & 31) << 2;
      const float4 v = *(const float4*)(Wf + (size_t)(k0 + r) * DOUT_ + c);
      Bs[r][c] = v.x; Bs[r][c + 1] = v.y; Bs[r][c + 2] = v.z; Bs[r][c + 3] = v.w;
    }
    __syncthreads();
#pragma unroll
    for (int kk = 0; kk < 16; kk += 4) {
      v2f a, b;
      a.x = As[m][kk + kb];       a.y = As[m][kk + kb + 1];
      b.x = Bs[kk + kb][n0 + m];  b.y = Bs[kk + kb + 1][n0 + m];
      acc = __builtin_amdgcn_wmma_f32_16x16x4_f32(false, a, false, b,
                                                  (short)0, acc, false, false);
    }
  }
  const int col = n0 + m;
  const int mhi = (lane >> 4) << 3;
  const float bv = bf[col];
#pragma unroll
  for (int r = 0; r < 8; ++r) {
    int i = row0 + r + mhi;               // global row = t*BN + j
    int t = i / BN_;
    int j = i - t * BN_;
    int b = j / NN_;
    int node = j - b * NN_;
    float v = (acc[r] + bv) * fm[i];
    out[(size_t)((b * NN_ + node) * T_ + t) * DOUT_ + col] = v;
  }
}

// ---------------------------------------------------------------------------
extern "C" void kernel_launch(void* const* d_in, const int* in_sizes, int n_in,
                              void* d_out, int out_size, void* d_ws, size_t ws_size,
                              hipStream_t stream) {
  (void)in_sizes; (void)n_in; (void)out_size; (void)ws_size;
  const float* x    = (const float*)d_in[0];
  const float* A    = (const float*)d_in[1];
  const int*   mask = (const int*)d_in[2];
  const float* W1   = (const float*)d_in[3];
  const float* b1   = (const float*)d_in[4];
  const float* W2   = (const float*)d_in[5];
  const float* b2   = (const float*)d_in[6];
  const float* Wf   = (const float*)d_in[7];
  const float* bf   = (const float*)d_in[8];
  float* out = (float*)d_out;

  float* ws   = (float*)d_ws;
  float* fm   = ws;                                  // 40000 floats
  float* dinv = ws + 40000;                          // 40000 floats
  float* bufA = ws + 80000;                          // 10.24M floats
  float* bufB = bufA + (size_t)T_ * BN_ * DH_;       // 10.24M floats

  k_mask<<<(B_ * T_ * NN_ + 255) / 256, 256, 0, stream>>>(mask, fm);
  k_dinv<<<(T_ * BN_ + 7) / 8, 256, 0, stream>>>(A, fm, dinv);
  k_hw1<<<T_ * BN_, DH_, 0, stream>>>(x, W1, dinv, bufA);   // hw1 -> bufA
  dim3 gp(BN_ / 16, T_);
  k_prop<<<gp, 256, 0, stream>>>(A, bufA, dinv, b1, bufB);  // h1 -> bufB
  k_mm_scale<<<(T_ * BN_) / 16, 256, 0, stream>>>(bufB, W2, dinv, bufA); // hw2 -> bufA
  k_prop<<<gp, 256, 0, stream>>>(A, bufA, dinv, b2, bufB);  // h2 -> bufB
  k_final<<<(T_ * BN_) / 16, 256, 0, stream>>>(bufB, Wf, bf, fm, out);
}